// TripletLoss_6021544149806
// MI455X (gfx1250) — compile-verified
//
#include <hip/hip_runtime.h>
#include <math.h>

typedef float v2f __attribute__((ext_vector_type(2)));
typedef float v8f __attribute__((ext_vector_type(8)));

#define BATCH   4096
#define DIM     1024
#define K4STEPS (DIM / 4)      // 256 wmma k-steps of 4
#define NTILES  (BATCH / 16)   // 256 column tiles
#define MROWS   32             // rows per block (two 16-row WMMA sub-tiles)
#define LDSTR   1028           // row stride in floats: 1024 + 4 pad (16B aligned, bank-conflict free)
#define LARGE_F 1e5f

// ---------------------------------------------------------------------------
// Kernel 1: per-row squared norms. 8 waves/block, one wave per row.
// ---------------------------------------------------------------------------
__global__ void sqnorm_kernel(const float* __restrict__ feat,
                              float* __restrict__ sq) {
    const int wave = threadIdx.x >> 5;
    const int lane = threadIdx.x & 31;
    const int row  = blockIdx.x * 8 + wave;
    const float4* fr = (const float4*)(feat + (size_t)row * DIM);

    float s = 0.0f;
    #pragma unroll
    for (int i = 0; i < 8; ++i) {              // 8 * 32 lanes * 4 floats = 1024
        float4 v = fr[lane + i * 32];
        s += v.x * v.x + v.y * v.y + v.z * v.z + v.w * v.w;
    }
    #pragma unroll
    for (int off = 16; off > 0; off >>= 1)
        s += __shfl_xor(s, off, 32);
    if (lane == 0) sq[row] = s;
}

// ---------------------------------------------------------------------------
// Kernel 2: batch-hard triplet loss via fp32 WMMA Gram tiles.
// Block = 256 threads (8 waves). Block b owns rows [32b, 32b+32); each global
// B fragment feeds TWO wmma ops (rows 0-15 and 16-31 sub-tiles from LDS).
// ---------------------------------------------------------------------------
__global__ void triplet_kernel(const float* __restrict__ feat,
                               const int*   __restrict__ ids,
                               const float* __restrict__ sq,
                               float*       __restrict__ out) {
    extern __shared__ float smem[];
    float* ldsA   = smem;                          // MROWS * LDSTR floats
    float* redMax = smem + MROWS * LDSTR;          // 8 waves * 32 rows
    float* redMin = redMax + 8 * MROWS;            // 8 waves * 32 rows

    const int tid    = threadIdx.x;
    const int wave   = tid >> 5;
    const int lane   = tid & 31;
    const int lane15 = lane & 15;
    const int half   = lane >> 4;
    const int rowBase = blockIdx.x * MROWS;

    // Stage the 32-row A tile into padded LDS (ds_store_b128, conflict-free).
    {
        const float4* src = (const float4*)(feat + (size_t)rowBase * DIM);
        #pragma unroll
        for (int i = 0; i < 2 * 16; ++i) {         // 8192 float4s / 256 threads
            int idx = i * 256 + tid;
            int r   = idx >> 8;                    // row 0..31
            int k4  = idx & 255;                   // float4 index within row
            float4 v = src[r * (DIM / 4) + k4];
            *(float4*)(&ldsA[r * LDSTR + k4 * 4]) = v;
        }
    }
    __syncthreads();

    // Per-lane row-side constants for both sub-tiles:
    //   set 0: m = rowBase + v + 8*half ; set 1: m + 16
    float sq_m0[8], sq_m1[8];
    int   id_m0[8], id_m1[8];
    #pragma unroll
    for (int v = 0; v < 8; ++v) {
        int m0 = rowBase + v + 8 * half;
        sq_m0[v] = sq[m0];      id_m0[v] = ids[m0];
        sq_m1[v] = sq[m0 + 16]; id_m1[v] = ids[m0 + 16];
    }

    float maxp0[8], minn0[8], maxp1[8], minn1[8];
    #pragma unroll
    for (int v = 0; v < 8; ++v) {
        maxp0[v] = 0.0f; minn0[v] = 3.402823466e38f;
        maxp1[v] = 0.0f; minn1[v] = 3.402823466e38f;
    }

    // A fragment sources: row = lane%16 (+16 for sub-tile 1), K offset 2*half.
    const float* aptr0 = &ldsA[lane15 * LDSTR + 2 * half];
    const float* aptr1 = aptr0 + 16 * LDSTR;

    // Each wave sweeps 32 disjoint 16-column tiles (uniform trip count -> EXEC all ones).
    for (int ct = wave; ct < NTILES; ct += 8) {
        const int   colBase = ct * 16;
        const int   n       = colBase + lane15;
        const float sq_n    = sq[n];
        const int   id_n    = ids[n];
        // B fragment source: column row n of feat, K offset 2*half.
        const float* bptr = feat + (size_t)n * DIM + 2 * half;

        v8f c0 = {};
        v8f c1 = {};
        #pragma unroll 4
        for (int k4 = 0; k4 < K4STEPS; ++k4) {
            v2f b  = *(const v2f*)(bptr  + k4 * 4);   // global_load_b64 (L2-resident)
            v2f a0 = *(const v2f*)(aptr0 + k4 * 4);   // ds_load_b64
            v2f a1 = *(const v2f*)(aptr1 + k4 * 4);   // ds_load_b64
            c0 = __builtin_amdgcn_wmma_f32_16x16x4_f32(
                     false, a0, false, b, (short)0, c0, false, false);
            c1 = __builtin_amdgcn_wmma_f32_16x16x4_f32(
                     false, a1, false, b, (short)0, c1, false, false);
        }

        // Epilogue: distances + batch-hard updates for this 32x16 tile.
        #pragma unroll
        for (int v = 0; v < 8; ++v) {
            int   m0 = rowBase + v + 8 * half;
            float d2 = sq_m0[v] + sq_n - 2.0f * c0[v];
            float d  = sqrtf(fmaxf(d2, 0.0f) + 1e-12f);
            bool same = (id_m0[v] == id_n);
            if (same && (m0 != n)) maxp0[v] = fmaxf(maxp0[v], d);
            minn0[v] = fminf(minn0[v], same ? d + LARGE_F : d);

            int   m1  = m0 + 16;
            float e2  = sq_m1[v] + sq_n - 2.0f * c1[v];
            float e   = sqrtf(fmaxf(e2, 0.0f) + 1e-12f);
            bool same1 = (id_m1[v] == id_n);
            if (same1 && (m1 != n)) maxp1[v] = fmaxf(maxp1[v], e);
            minn1[v] = fminf(minn1[v], same1 ? e + LARGE_F : e);
        }
    }

    // Butterfly reduce across the 16 lanes that share each row (offsets < 16
    // never cross the half boundary, so width 32 is safe).
    #pragma unroll
    for (int off = 1; off < 16; off <<= 1) {
        #pragma unroll
        for (int v = 0; v < 8; ++v) {
            maxp0[v] = fmaxf(maxp0[v], __shfl_xor(maxp0[v], off, 32));
            minn0[v] = fminf(minn0[v], __shfl_xor(minn0[v], off, 32));
            maxp1[v] = fmaxf(maxp1[v], __shfl_xor(maxp1[v], off, 32));
            minn1[v] = fminf(minn1[v], __shfl_xor(minn1[v], off, 32));
        }
    }
    if (lane15 == 0) {
        #pragma unroll
        for (int v = 0; v < 8; ++v) {
            int mloc = v + 8 * half;
            redMax[wave * MROWS + mloc]      = maxp0[v];
            redMin[wave * MROWS + mloc]      = minn0[v];
            redMax[wave * MROWS + mloc + 16] = maxp1[v];
            redMin[wave * MROWS + mloc + 16] = minn1[v];
        }
    }
    __syncthreads();

    // Cross-wave reduction + soft-margin, one thread per output row.
    if (tid < MROWS) {
        float mp = 0.0f, mn = 3.402823466e38f;
        #pragma unroll
        for (int w = 0; w < 8; ++w) {
            mp = fmaxf(mp, redMax[w * MROWS + tid]);
            mn = fminf(mn, redMin[w * MROWS + tid]);
        }
        float z  = mp - mn;
        float sp = fmaxf(z, 0.0f) + log1pf(expf(-fabsf(z)));   // stable softplus
        out[rowBase + tid] = sp;
    }
}

// ---------------------------------------------------------------------------
extern "C" void kernel_launch(void* const* d_in, const int* in_sizes, int n_in,
                              void* d_out, int out_size, void* d_ws, size_t ws_size,
                              hipStream_t stream) {
    const float* feat = (const float*)d_in[0];
    const int*   ids  = (const int*)d_in[1];
    float*       out  = (float*)d_out;
    float*       sq   = (float*)d_ws;            // 4096 floats of scratch

    (void)in_sizes; (void)n_in; (void)out_size; (void)ws_size;

    // 1) row squared norms: 512 blocks x 256 threads (one wave per row)
    sqnorm_kernel<<<BATCH / 8, 256, 0, stream>>>(feat, sq);

    // 2) main pass: 128 blocks (32 rows each) x 256 threads,
    //    dynamic LDS = padded 32-row A tile + reduction buffers (133,632 B)
    size_t shmem = (size_t)(MROWS * LDSTR + 2 * 8 * MROWS) * sizeof(float);
    triplet_kernel<<<BATCH / MROWS, 256, shmem, stream>>>(feat, ids, sq, out);
}